// GroupAndAlign_43817256354257
// MI455X (gfx1250) — compile-verified
//
#include <hip/hip_runtime.h>
#include <hip/hip_bf16.h>

// Problem constants (from reference setup_inputs)
#define B_    4
#define NPC   40000
#define MBOX  256
#define NS    512
#define R2    1.0f
#define CTILES (NPC / 32)   // 1250, exact: 32 points per iteration

typedef __attribute__((ext_vector_type(2))) float v2f;
typedef __attribute__((ext_vector_type(8))) float v8f;

// ---------------------------------------------------------------------------
// Kernel 1: ball query via WMMA distance tiles, 32 points / iteration.
// One wave (32 threads) per (batch, 16-center tile).
//   d2[m][n] = |c_m|^2 - 2 c_m . p_n + |p_n|^2
// computed as D = A x B + C with V_WMMA_F32_16X16X4_F32 (two per iteration):
//   A (16x4): row m = (-2cx, -2cy, -2cz, 1)   lanes 0-15: K=0,1; lanes 16-31: K=2,3
//   B (4x16): col n = (px, py, pz, |p|^2)     lanes 0-15: K=0,1 rows; lanes 16-31: K=2,3
//   C (16x16): row m broadcast |c_m|^2        VGPR r: rows r (lanes 0-15) / r+8 (16-31)
// Each lane loads ONE point (t*32+lane); the two wave halves swap missing
// components with 4 shuffles to build both B operands. Per D VGPR, one v_cmp
// yields two 16-bit row masks as a uniform SGPR; the 16 words (2 tiles x 8)
// are stored interleaved in LDS so lane m fetches both of its masks with one
// ds_load_b64, forming a 32-bit ascending-index mask for the ordered walk.
// ---------------------------------------------------------------------------
__global__ __launch_bounds__(32) void BallQueryWmma_kernel(
    const float* __restrict__ boxXyz,   // (B, MBOX, 3)
    const float* __restrict__ pc,       // (B, NPC, 7)
    int* __restrict__ idxOut)           // (B, MBOX, NS)
{
    __shared__ __align__(16) unsigned sb[16];   // interleaved: {bl_r tile0, bl_r tile1} pairs

    const int tile = blockIdx.x;                 // 0 .. B*MBOX/16 - 1
    const int b    = tile / (MBOX / 16);
    const int ct0  = (tile % (MBOX / 16)) * 16;  // first center of this tile
    const int lane = threadIdx.x;                // 0..31 (wave32)
    const int lm   = lane & 15;
    const bool hi  = lane >= 16;

    // --- Load my center (center index ct0+lm), build A and C operands ------
    const float* cptr = boxXyz + ((size_t)b * MBOX + (ct0 + lm)) * 3;
    const float cx = cptr[0], cy = cptr[1], cz = cptr[2];
    const float cn = cx * cx + cy * cy + cz * cz;

    v2f a;
    a.x = hi ? (-2.0f * cz) : (-2.0f * cx);   // K=2 : K=0
    a.y = hi ? (1.0f)       : (-2.0f * cy);   // K=3 : K=1

    v8f cacc;
#pragma unroll
    for (int r = 0; r < 8; ++r)               // C VGPR r = |c_row|^2, row = r (+8 for hi lanes)
        cacc[r] = __shfl(cn, r + (hi ? 8 : 0), 32);

    int cnt = 0;
    int firstIdx = 0;
    const float* pbase = pc + (size_t)b * NPC * 7;
    int* myOut = idxOut + ((size_t)b * MBOX + (ct0 + lm)) * NS;  // meaningful for lanes 0-15

    // Software pipeline: preload my point of tile 0, prefetch tile 1.
    const float* pp0 = pbase + (size_t)lane * 7;
    __builtin_prefetch(pp0 + 32 * 7, 0, 1);
    float px = pp0[0], py = pp0[1], pz = pp0[2];

    for (int t = 0; t < CTILES; ++t) {
        const float pn = px * px + py * py + pz * pz;

        // Build both B operands: tile0 = points t*32+0..15, tile1 = +16..31.
        const float zlo = __shfl(pz, lm, 32);        // point lm's z (for hi lanes, tile0)
        const float nlo = __shfl(pn, lm, 32);        // point lm's |p|^2
        const float xhi = __shfl(px, lm + 16, 32);   // point lm+16's x (for lo lanes, tile1)
        const float yhi = __shfl(py, lm + 16, 32);   // point lm+16's y

        v2f b0, b1;
        b0.x = hi ? zlo : px;   b0.y = hi ? nlo : py;
        b1.x = hi ? pz  : xhi;  b1.y = hi ? pn  : yhi;

        // Two independent WMMAs sharing A and C (EXEC all-ones: uniform control flow).
        v8f d0 = __builtin_amdgcn_wmma_f32_16x16x4_f32(
            false, a, false, b0, (short)0, cacc, false, false);
        v8f d1 = __builtin_amdgcn_wmma_f32_16x16x4_f32(
            false, a, false, b1, (short)0, cacc, false, false);

        // Issue next tile's load now; its wait lands after the selection walk.
        if (t + 1 < CTILES) {
            const float* np = pbase + (size_t)((t + 1) * 32 + lane) * 7;
            __builtin_prefetch(np + 32 * 7, 0, 1);   // prefetch tile t+2
            px = np[0]; py = np[1]; pz = np[2];
        }

        // Uniform masks: bits[15:0] of each ballot = row r, bits[31:16] = row r+8.
        const unsigned a0 = (unsigned)__ballot(d0[0] < R2);
        const unsigned a1 = (unsigned)__ballot(d0[1] < R2);
        const unsigned a2 = (unsigned)__ballot(d0[2] < R2);
        const unsigned a3 = (unsigned)__ballot(d0[3] < R2);
        const unsigned a4 = (unsigned)__ballot(d0[4] < R2);
        const unsigned a5 = (unsigned)__ballot(d0[5] < R2);
        const unsigned a6 = (unsigned)__ballot(d0[6] < R2);
        const unsigned a7 = (unsigned)__ballot(d0[7] < R2);
        const unsigned c0 = (unsigned)__ballot(d1[0] < R2);
        const unsigned c1 = (unsigned)__ballot(d1[1] < R2);
        const unsigned c2 = (unsigned)__ballot(d1[2] < R2);
        const unsigned c3 = (unsigned)__ballot(d1[3] < R2);
        const unsigned c4 = (unsigned)__ballot(d1[4] < R2);
        const unsigned c5 = (unsigned)__ballot(d1[5] < R2);
        const unsigned c6 = (unsigned)__ballot(d1[6] < R2);
        const unsigned c7 = (unsigned)__ballot(d1[7] < R2);

        // Interleaved store: pair 2r = (tile0 bl_r, tile1 bl_r) -> one ds_load_b64/lane.
        if (lane == 0) {
            ((uint4*)sb)[0] = make_uint4(a0, c0, a1, c1);
            ((uint4*)sb)[1] = make_uint4(a2, c2, a3, c3);
            ((uint4*)sb)[2] = make_uint4(a4, c4, a5, c5);
            ((uint4*)sb)[3] = make_uint4(a6, c6, a7, c7);
        }
        __syncthreads();   // single-wave WG: degenerates to DS-count ordering

        const uint2 w = ((const uint2*)sb)[lane & 7];
        const unsigned m0 = (lane < 8) ? (w.x & 0xFFFFu) : (w.x >> 16);
        const unsigned m1 = (lane < 8) ? (w.y & 0xFFFFu) : (w.y >> 16);
        unsigned rowmask = (lane < 16) ? (m0 | (m1 << 16)) : 0u;

        while (rowmask && cnt < NS) {         // ordered: ctz walks ascending point index
            const int bit = __builtin_ctz(rowmask);
            rowmask &= rowmask - 1;
            const int pj = t * 32 + bit;
            if (cnt == 0) firstIdx = pj;
            myOut[cnt++] = pj;
        }

        // Early exit once every center in the tile has NS neighbors.
        const bool done = (lane < 16) ? (cnt >= NS) : true;
        if (__ballot(done) == 0xFFFFFFFFull) break;
    }

    // Pad remaining slots with the first found index (0 if none) — reference semantics.
    if (lane < 16) {
        const int f = (cnt > 0) ? firstIdx : 0;
        for (int s = cnt; s < NS; ++s) myOut[s] = f;
    }
}

// ---------------------------------------------------------------------------
// Kernel 2: gather + z-rotate + transposed writes.
// One thread per (b, m, s). All output writes are contiguous across lanes:
//   out_xyz : (MBOX, B, NS, 3)               [first 1,572,864 floats]
//   g_feat  : (B, 5, MBOX, NS)               [next  2,621,440 floats]
// ---------------------------------------------------------------------------
__global__ __launch_bounds__(256) void GatherRotate_kernel(
    const float* __restrict__ boxXyz,   // (B, MBOX, 3)
    const float* __restrict__ ori,      // (B, MBOX)
    const float* __restrict__ pc,       // (B, NPC, 7)
    const float* __restrict__ labels,   // (B, NPC)
    const int*   __restrict__ idx,      // (B, MBOX, NS)
    float* __restrict__ out)
{
    const int t  = blockIdx.x * 256 + threadIdx.x;   // over B*MBOX*NS
    const int s  = t & (NS - 1);
    const int bm = t >> 9;                            // NS == 512
    const int m  = bm % MBOX;
    const int b  = bm / MBOX;

    const int j = idx[(size_t)bm * NS + s];
    const float* pp = pc     + ((size_t)b * NPC + j) * 7;
    const float* cp = boxXyz + ((size_t)b * MBOX + m) * 3;

    float si, co;
    __sincosf(ori[(size_t)b * MBOX + m], &si, &co);

    const float x = pp[0] - cp[0];
    const float y = pp[1] - cp[1];
    const float z = pp[2] - cp[2];
    const float rx = (co * x + si * y);   // / RADIUS == 1.0
    const float ry = (co * y - si * x);

    // out_xyz[m][b][s][k] — 3 consecutive floats per thread, contiguous per wave
    const size_t xo = (((size_t)m * B_ + b) * NS + s) * 3;
    out[xo + 0] = rx;
    out[xo + 1] = ry;
    out[xo + 2] = z;

    // g_feat[b][f][m][s] — each plane coalesced along s
    const size_t X      = (size_t)MBOX * B_ * NS * 3;
    const size_t stride = (size_t)MBOX * NS;
    const size_t fb     = X + ((size_t)b * 5 * MBOX + m) * NS + s;
    out[fb + 0 * stride] = pp[3];
    out[fb + 1 * stride] = pp[4];
    out[fb + 2 * stride] = pp[5];
    out[fb + 3 * stride] = pp[6];
    out[fb + 4 * stride] = labels[(size_t)b * NPC + j];
}

extern "C" void kernel_launch(void* const* d_in, const int* in_sizes, int n_in,
                              void* d_out, int out_size, void* d_ws, size_t ws_size,
                              hipStream_t stream) {
    const float* boxXyz = (const float*)d_in[0];  // (B, MBOX, 3)
    const float* ori    = (const float*)d_in[1];  // (B, MBOX)
    // d_in[2] box_feature: unused by reference outputs
    const float* pc     = (const float*)d_in[3];  // (B, NPC, 7)
    const float* labels = (const float*)d_in[4];  // (B, NPC)
    // d_in[5] proposal_instance_labels: unused

    int* idxWs = (int*)d_ws;                      // B*MBOX*NS int32 = 2 MB scratch

    BallQueryWmma_kernel<<<B_ * MBOX / 16, 32, 0, stream>>>(boxXyz, pc, idxWs);
    GatherRotate_kernel<<<B_ * MBOX * NS / 256, 256, 0, stream>>>(
        boxXyz, ori, pc, labels, idxWs, (float*)d_out);
}